// LayerTokenPrune_32220844655195
// MI455X (gfx1250) — compile-verified
//
#include <hip/hip_runtime.h>
#include <hip/hip_bf16.h>
#include <stdint.h>

// Problem constants (fixed shapes from the reference)
#define B_    32
#define C_    192
#define L_    4096
#define ALPHA_ 0.1f

typedef __attribute__((ext_vector_type(2))) float v2f;
typedef __attribute__((ext_vector_type(4))) float f4;
typedef __attribute__((ext_vector_type(8))) float v8f;
typedef __attribute__((ext_vector_type(4))) int v4i;

#if defined(__gfx1250__) && __has_builtin(__builtin_amdgcn_wmma_f32_16x16x4_f32)
#define HAVE_WMMA 1
#else
#define HAVE_WMMA 0
#endif

#if defined(__gfx1250__) && __has_builtin(__builtin_amdgcn_global_load_async_to_lds_b128)
#define HAVE_ASYNC 1
#else
#define HAVE_ASYNC 0
#endif

#if HAVE_ASYNC
// Param types per hipcc diagnostic: v4i in global (AS1) / LDS (AS3) space.
typedef __attribute__((address_space(1))) v4i gv4i;
typedef __attribute__((address_space(3))) v4i lv4i;
static __device__ inline void wait_async0() {
#if __has_builtin(__builtin_amdgcn_s_wait_asynccnt)
  __builtin_amdgcn_s_wait_asynccnt(0);
#else
  asm volatile("s_wait_asynccnt 0" ::: "memory");
#endif
}
#endif

// ---- wave32 reductions -----------------------------------------------------
__device__ inline float wave_sum(float v) {
#if HAVE_WMMA
  // D = A x ones + 0 : row-sum of A. A is 16x4 f32 striped over 32 lanes
  // (lane m / m+16 hold K=0..1 / K=2..3). Put the partial in A[:,0]/A[:,2],
  // zeros in the other K slots; every D element of row m equals p[m]+p[m+16].
  v2f a; a.x = v; a.y = 0.0f;
  v2f bm; bm.x = 1.0f; bm.y = 1.0f;   // B = all ones (layout-independent)
  v8f c = {};
  c = __builtin_amdgcn_wmma_f32_16x16x4_f32(false, a, false, bm,
                                            (short)0, c, false, false);
  float s = c[0] + c[1] + c[2] + c[3] + c[4] + c[5] + c[6] + c[7];
  // lanes 0-15 hold sum over M=0..7, lanes 16-31 over M=8..15
  s += __shfl_xor(s, 16, 32);
  return s;  // all lanes: full 32-lane sum
#else
#pragma unroll
  for (int off = 16; off >= 1; off >>= 1) v += __shfl_xor(v, off, 32);
  return v;
#endif
}

__device__ inline float block_sum(float x, float* s_w, int tid) {
  float w = wave_sum(x);
  __syncthreads();                       // protect s_w reuse
  if ((tid & 31) == 0) s_w[tid >> 5] = w;
  __syncthreads();
  float t = 0.0f;
#pragma unroll
  for (int i = 0; i < 8; ++i) t += s_w[i];  // same order in all threads
  return t;
}

// ---- kernel 1: per-batch stats + stable compaction permutation -------------
__global__ __launch_bounds__(256) void prune_stats_kernel(
    const float* __restrict__ delta, int* __restrict__ srcIdx,
    int* __restrict__ counts, float* __restrict__ thetas) {
  __shared__ __align__(16) float s_imp[L_];
  __shared__ float s_w[16];
  __shared__ int s_scan[256];

  const int tid = threadIdx.x;
  const int b = blockIdx.x;
  const float* gsrc = delta + (size_t)b * L_;

  // Stage the 16KB delta tile into LDS (CDNA5 async copy when available).
#if HAVE_ASYNC
#pragma unroll
  for (int i = 0; i < 4; ++i) {
    int chunk = tid + i * 256;  // one float4 per async op
    __builtin_amdgcn_global_load_async_to_lds_b128(
        (gv4i*)(gsrc + (size_t)chunk * 4),
        (lv4i*)(&s_imp[chunk * 4]), 0, 0);
  }
  wait_async0();
#else
#pragma unroll
  for (int i = 0; i < 4; ++i) {
    int chunk = tid + i * 256;
    ((f4*)s_imp)[chunk] = ((const f4*)gsrc)[chunk];
  }
#endif
  __syncthreads();

  float v[16];
#pragma unroll
  for (int j = 0; j < 16; ++j) v[j] = __builtin_fabsf(s_imp[tid * 16 + j]);

  // min / max over the batch
  float lo = v[0], hi = v[0];
#pragma unroll
  for (int j = 1; j < 16; ++j) { lo = fminf(lo, v[j]); hi = fmaxf(hi, v[j]); }
#pragma unroll
  for (int off = 16; off >= 1; off >>= 1) {
    lo = fminf(lo, __shfl_xor(lo, off, 32));
    hi = fmaxf(hi, __shfl_xor(hi, off, 32));
  }
  if ((tid & 31) == 0) { s_w[tid >> 5] = lo; s_w[8 + (tid >> 5)] = hi; }
  __syncthreads();
  float lo_all = s_w[0], hi_all = s_w[8];
#pragma unroll
  for (int i = 1; i < 8; ++i) {
    lo_all = fminf(lo_all, s_w[i]);
    hi_all = fmaxf(hi_all, s_w[8 + i]);
  }
  __syncthreads();
  const float rng = fmaxf(hi_all - lo_all, 1e-3f);

  // normalize (exact elementwise order as reference: (x-lo)/rng)
#pragma unroll
  for (int j = 0; j < 16; ++j) v[j] = (v[j] - lo_all) / rng;

  // mean
  float ls = 0.0f;
#pragma unroll
  for (int j = 0; j < 16; ++j) ls += v[j];
  const float mu = block_sum(ls, s_w, tid) * (1.0f / (float)L_);

  // unbiased std (two-pass: avoids fp32 cancellation)
  float ld = 0.0f;
#pragma unroll
  for (int j = 0; j < 16; ++j) { float d = v[j] - mu; ld += d * d; }
  const float sigma = sqrtf(block_sum(ld, s_w, tid) / (float)(L_ - 1));
  const float theta = mu - ALPHA_ * sigma;

  // keep mask + stable compaction: kept indices first, dropped at tail
  unsigned km = 0u;
  int kc = 0;
#pragma unroll
  for (int j = 0; j < 16; ++j)
    if (v[j] >= theta) { km |= (1u << j); ++kc; }

  s_scan[tid] = kc;
  __syncthreads();
  for (int off = 1; off < 256; off <<= 1) {
    int a = s_scan[tid];
    int addv = (tid >= off) ? s_scan[tid - off] : 0;
    __syncthreads();
    s_scan[tid] = a + addv;
    __syncthreads();
  }
  const int incl = s_scan[tid];
  const int total = s_scan[255];
  int kpos = incl - kc;                 // exclusive prefix of keeps
  int dpos = total + (tid * 16 - kpos); // drops go after all keeps, in order
  int* dst = srcIdx + (size_t)b * L_;
#pragma unroll
  for (int j = 0; j < 16; ++j) {
    int l = tid * 16 + j;
    if (km & (1u << j)) dst[kpos++] = l;
    else                dst[dpos++] = l;
  }
  if (tid == 0) { counts[b] = total; thetas[b] = theta; }
}

// ---- kernel 2: bandwidth kernel — gather x along L, zero-pad tail ----------
__global__ __launch_bounds__(256) void gather_kernel(
    const float* __restrict__ x, const int* __restrict__ srcIdx,
    const int* __restrict__ counts, float* __restrict__ y) {
  const int b = blockIdx.x >> 2;          // 4 L-tiles per batch
  const int tile = blockIdx.x & 3;
  const int l0 = tile * 1024 + threadIdx.x * 4;
  const int cnt = counts[b];
  const int* si = srcIdx + (size_t)b * L_;
  const int s0 = si[l0 + 0], s1 = si[l0 + 1], s2 = si[l0 + 2], s3 = si[l0 + 3];
  const float m0 = (l0 + 0 < cnt) ? 1.0f : 0.0f;
  const float m1 = (l0 + 1 < cnt) ? 1.0f : 0.0f;
  const float m2 = (l0 + 2 < cnt) ? 1.0f : 0.0f;
  const float m3 = (l0 + 3 < cnt) ? 1.0f : 0.0f;

  const int c0 = blockIdx.y * 48;         // 4 channel groups of 48
  const float* xb = x + ((size_t)b * C_ + c0) * L_;
  float* yb = y + ((size_t)b * C_ + c0) * L_ + l0;

#pragma unroll 4
  for (int c = 0; c < 48; ++c) {
    const float* xc = xb + (size_t)c * L_;
    f4 o;
    o.x = xc[s0] * m0;
    o.y = xc[s1] * m1;
    o.z = xc[s2] * m2;
    o.w = xc[s3] * m3;
    // y is write-once: non-temporal store keeps L2 for the gather reads
    __builtin_nontemporal_store(o, (f4*)(yb + (size_t)c * L_));
  }
}

// ---- kernel 3: scalars -----------------------------------------------------
__global__ void finalize_kernel(const int* __restrict__ counts,
                                const float* __restrict__ thetas,
                                float* __restrict__ out) {
  const int t = threadIdx.x;  // 32 threads == one wave
  float c = (float)counts[t];
  float th = thetas[t];
#pragma unroll
  for (int off = 16; off >= 1; off >>= 1) {
    c += __shfl_xor(c, off, 32);
    th += __shfl_xor(th, off, 32);
  }
  if (t == 0) {
    out[0] = (c / (float)B_) / (float)L_;  // keep_ratio
    out[1] = th / (float)B_;               // theta_mean
  }
}

extern "C" void kernel_launch(void* const* d_in, const int* in_sizes, int n_in,
                              void* d_out, int out_size, void* d_ws,
                              size_t ws_size, hipStream_t stream) {
  (void)in_sizes; (void)n_in; (void)out_size; (void)ws_size;
  const float* x = (const float*)d_in[0];
  const float* delta = (const float*)d_in[1];
  float* y = (float*)d_out;

  int* srcIdx = (int*)d_ws;                                   // 512 KB
  int* counts = (int*)((char*)d_ws + (size_t)B_ * L_ * 4);    // 128 B
  float* thetas = (float*)((char*)d_ws + (size_t)B_ * L_ * 4 + B_ * 4);

  prune_stats_kernel<<<B_, 256, 0, stream>>>(delta, srcIdx, counts, thetas);
  gather_kernel<<<dim3(B_ * 4, 4), 256, 0, stream>>>(x, srcIdx, counts, y);
  finalize_kernel<<<1, 32, 0, stream>>>(counts, thetas,
                                        y + (size_t)B_ * C_ * L_);
}